// Inference_and_Generation_20220706030061
// MI455X (gfx1250) — compile-verified
//
#include <hip/hip_runtime.h>

// ---------------- problem constants (fixed by setup_inputs) ----------------
#define BB 8
#define WDIM 256
#define HDIM 256
#define NWC 32
#define NHC 32
#define NCELL (NWC*NHC)          // 1024
#define DZW 4
#define DZI 16
#define FCH 32
#define SS 28
#define KOBJ 25
#define KB (KOBJ*BB)             // 200
#define FEAT_FLAT (FCH*SS*SS)    // 25088
#define OUT_FLAT (2*SS*SS)       // 1568
#define SIZE_MIN 24.0f
#define SIZE_MAX 96.0f
#define KCHUNKS 32
#define KPER (FEAT_FLAT/KCHUNKS) // 784
#define MTILES 13                // ceil(200/16)
#define MPAD (MTILES*16)         // 208

// ---------------- output offsets (floats, reference return order) ----------
static constexpr size_t O_LOGIT  = 0;                      // (8,1,32,32)   8192
static constexpr size_t O_LOGITF = 8192;                   // (25,8)        200
static constexpr size_t O_BMASK  = 8392;                   // (25,8,1,256,256)
static constexpr size_t O_BMASKN = O_BMASK  + 13107200;
static constexpr size_t O_BIMG   = O_BMASKN + 13107200;
static constexpr size_t O_PMAP   = O_BIMG   + 13107200;    // 8192
static constexpr size_t O_CFEW   = O_PMAP + 8192;          // 200
static constexpr size_t O_BX     = O_CFEW + 200;
static constexpr size_t O_BY     = O_BX + 200;
static constexpr size_t O_BW     = O_BY + 200;
static constexpr size_t O_BH     = O_BW + 200;
static constexpr size_t O_ZS     = O_BH + 200;             // (25,8,16) 3200
static constexpr size_t O_KLL    = O_ZS + 3200;            // 8192
static constexpr size_t O_KLZW   = O_KLL + 8192;           // (25,8,4)  800
static constexpr size_t O_KLZI   = O_KLZW + 800;           // 3200  -> total 39354576

// ---------------- workspace offsets (floats) -------------------------------
static constexpr size_t W_BOXES = 0;                                    // 800
static constexpr size_t W_ZPART = 800;                                  // 2*32*208*16
static constexpr size_t W_ZMU   = W_ZPART + (size_t)2*KCHUNKS*MPAD*DZI; // 3200
static constexpr size_t W_SMALL = W_ZMU + (size_t)KB*DZI;               // 313600
static constexpr size_t W_XF    = W_SMALL + (size_t)KB*OUT_FLAT;        // 5017600

typedef float v2f __attribute__((ext_vector_type(2)));
typedef float v8f __attribute__((ext_vector_type(8)));

__device__ __forceinline__ float sigmoidf_(float x){ return 1.f/(1.f+expf(-x)); }
__device__ __forceinline__ float softplusf_(float x){
  return x > 0.f ? x + log1pf(expf(-x)) : log1pf(expf(x));
}
__device__ __forceinline__ float tap2d(const float* __restrict__ im, int xi, int yi, int X, int Y){
  if (xi < 0 || xi >= X || yi < 0 || yi >= Y) return 0.f;
  return im[xi*Y + yi];
}
__device__ __forceinline__ float bilin2d(const float* __restrict__ im, float px, float py, int X, int Y){
  float x0f = floorf(px), y0f = floorf(py);
  int x0 = (int)x0f, y0 = (int)y0f;
  float wx = px - x0f, wy = py - y0f;
  float g00 = tap2d(im, x0,   y0,   X, Y);
  float g01 = tap2d(im, x0,   y0+1, X, Y);
  float g10 = tap2d(im, x0+1, y0,   X, Y);
  float g11 = tap2d(im, x0+1, y0+1, X, Y);
  return (1.f-wx)*((1.f-wy)*g00 + wy*g01) + wx*((1.f-wy)*g10 + wy*g11);
}

// ---------------- kernel 1: per-cell maps ----------------------------------
__global__ void k_maps(const float* __restrict__ logit, float* __restrict__ out){
  int i = blockIdx.x*blockDim.x + threadIdx.x;
  if (i >= BB*NCELL) return;
  float lg = logit[i];
  float p  = sigmoidf_(lg);
  out[O_LOGIT + i] = lg;
  out[O_PMAP  + i] = p;
  out[O_KLL   + i] = 0.6931471805599453f + p*logf(p + 1e-8f) + (1.f-p)*logf(1.f-p + 1e-8f);
}

// ---------------- kernel 2: stable top-k + box params ----------------------
__global__ void k_topk(const float* __restrict__ logit,
                       const float* __restrict__ zmu,
                       const float* __restrict__ zstd,
                       const float* __restrict__ w_zw,
                       const float* __restrict__ b_zw,
                       float* __restrict__ out,
                       float* __restrict__ boxes){
  int b = blockIdx.x;
  int lane = threadIdx.x;                    // wave32
  __shared__ float pv[NCELL];
  __shared__ int   sidx[KOBJ];
  __shared__ float sp[KOBJ];
  for (int j = lane; j < NCELL; j += 32)
    pv[j] = sigmoidf_(logit[b*NCELL + j]);
  __syncthreads();
  for (int r = 0; r < KOBJ; ++r) {
    float best = -1.f; int bi = 1 << 30;
    for (int j = lane; j < NCELL; j += 32) {   // first occurrence wins on ties
      float v = pv[j];
      if (v > best) { best = v; bi = j; }
    }
    for (int off = 16; off > 0; off >>= 1) {   // (max value, min index)
      float ov = __shfl_down(best, off, 32);
      int   oi = __shfl_down(bi,   off, 32);
      if (ov > best || (ov == best && oi < bi)) { best = ov; bi = oi; }
    }
    bi = __shfl(bi, 0, 32);
    if (lane == 0) { sidx[r] = bi; sp[r] = pv[bi]; pv[bi] = -1.f; }
    __syncthreads();
  }
  if (lane < KOBJ) {
    int n  = sidx[lane];
    int kb = lane*BB + b;
    out[O_CFEW   + kb] = sp[lane];
    out[O_LOGITF + kb] = logit[b*NCELL + n];
    int iw = n / NHC, ih = n % NHC;
    float zm[DZW], zs[DZW];
    #pragma unroll
    for (int c = 0; c < DZW; ++c) {
      zm[c] = zmu [((b*DZW + c)*NWC + iw)*NHC + ih];
      zs[c] = zstd[((b*DZW + c)*NWC + iw)*NHC + ih];
    }
    float t[4];
    #pragma unroll
    for (int o = 0; o < 4; ++o) {
      float s = b_zw[o];
      #pragma unroll
      for (int c = 0; c < DZW; ++c) s += w_zw[o*DZW + c]*zm[c];
      t[o] = sigmoidf_(s);
    }
    float bx = (float)WDIM*((float)iw + t[0])/(float)NWC;
    float by = (float)HDIM*((float)ih + t[1])/(float)NHC;
    float bw = SIZE_MIN + (SIZE_MAX - SIZE_MIN)*t[2];
    float bh = SIZE_MIN + (SIZE_MAX - SIZE_MIN)*t[3];
    out[O_BX + kb] = bx; out[O_BY + kb] = by;
    out[O_BW + kb] = bw; out[O_BH + kb] = bh;
    boxes[kb*4+0] = bx; boxes[kb*4+1] = by;
    boxes[kb*4+2] = bw; boxes[kb*4+3] = bh;
    #pragma unroll
    for (int c = 0; c < DZW; ++c)
      out[O_KLZW + kb*DZW + c] = 0.5f*(zs[c]*zs[c] + zm[c]*zm[c] - 1.f) - logf(zs[c]);
  }
}

// ---------------- kernel 3: bilinear crop -> xf (200 x 25088) --------------
__global__ void k_crop(const float* __restrict__ features,
                       const float* __restrict__ boxes,
                       float* __restrict__ xf){
  int kb = blockIdx.x;
  int b  = kb % BB;
  float cx = boxes[kb*4+0], cy = boxes[kb*4+1];
  float cw = boxes[kb*4+2], chh = boxes[kb*4+3];
  float x00 = cx - 0.5f*cw, y00 = cy - 0.5f*chh;
  const float inv = 1.f/(float)(SS-1);
  const float* img = features + (size_t)b*FCH*WDIM*HDIM;
  for (int e = threadIdx.x; e < FEAT_FLAT; e += blockDim.x) {
    int c   = e / (SS*SS);
    int rem = e - c*(SS*SS);
    int i   = rem / SS;
    int j   = rem - i*SS;
    float px = x00 + (float)i*inv*cw;
    float py = y00 + (float)j*inv*chh;
    xf[(size_t)kb*FEAT_FLAT + e] = bilin2d(img + (size_t)c*WDIM*HDIM, px, py, WDIM, HDIM);
  }
}

// ---------------- kernel 4: encoder GEMM via V_WMMA_F32_16X16X4_F32 --------
// D[m,n] = sum_k xf[m,k] * w_enc[n,k] ; M=200(pad 208), N=16, K=25088 split 32 ways.
__global__ void k_enc(const float* __restrict__ xf,
                      const float* __restrict__ wmu,
                      const float* __restrict__ wstd,
                      float* __restrict__ zpart){
  int mt   = blockIdx.x;
  int kc   = blockIdx.y;
  int lane = threadIdx.x;        // one wave32, EXEC all-ones throughout
  int half = lane >> 4;          // K sub-pair: 0 -> K{0,1}, 1 -> K{2,3}
  int l    = lane & 15;          // A: M row ; B: N col
  int row  = mt*16 + l;
  float msk = (row < KB) ? 1.f : 0.f;
  int rowc  = (row < KB) ? row : (KB-1);
  const float* arow  = xf   + (size_t)rowc*FEAT_FLAT;
  const float* bmrow = wmu  + (size_t)l*FEAT_FLAT;   // B[k][n] = w[n][k]
  const float* bsrow = wstd + (size_t)l*FEAT_FLAT;
  v8f accM = {}; v8f accS = {};
  int kbase = kc*KPER + half*2;
  for (int kk = 0; kk < KPER; kk += 4) {             // uniform trip count
    int k = kbase + kk;
    v2f a, bm, bs;
    a.x  = arow[k]*msk;  a.y  = arow[k+1]*msk;
    bm.x = bmrow[k];     bm.y = bmrow[k+1];
    bs.x = bsrow[k];     bs.y = bsrow[k+1];
    accM = __builtin_amdgcn_wmma_f32_16x16x4_f32(false, a, false, bm, (short)0, accM, false, false);
    accS = __builtin_amdgcn_wmma_f32_16x16x4_f32(false, a, false, bs, (short)0, accS, false, false);
  }
  #pragma unroll
  for (int r = 0; r < 8; ++r) {                      // C/D layout: M=r(+8 for hi lanes)
    int m = mt*16 + r + half*8;
    zpart[((size_t)kc*MPAD + m)*DZI + l] = accM[r];
    zpart[(size_t)KCHUNKS*MPAD*DZI + ((size_t)kc*MPAD + m)*DZI + l] = accS[r];
  }
}

// ---------------- kernel 5: reduce partials, softplus, KL ------------------
__global__ void k_post(const float* __restrict__ zpart,
                       const float* __restrict__ bmu,
                       const float* __restrict__ bstd,
                       float* __restrict__ out,
                       float* __restrict__ zmu_ws){
  int i = blockIdx.x*blockDim.x + threadIdx.x;
  if (i >= KB*DZI) return;
  int m = i / DZI, n = i - m*DZI;
  float smu = 0.f, sst = 0.f;
  for (int kc = 0; kc < KCHUNKS; ++kc) {
    smu += zpart[((size_t)kc*MPAD + m)*DZI + n];
    sst += zpart[(size_t)KCHUNKS*MPAD*DZI + ((size_t)kc*MPAD + m)*DZI + n];
  }
  float mu = smu + bmu[n];
  float sd = softplusf_(sst + bstd[n]);
  out[O_ZS   + i] = mu;                       // z_sample = z_mu
  out[O_KLZI + i] = 0.5f*(sd*sd + mu*mu - 1.f) - logf(sd + 1e-8f);
  zmu_ws[i] = mu;
}

// ---------------- kernel 6: decoder + activations --------------------------
__global__ void k_dec(const float* __restrict__ zmu_ws,
                      const float* __restrict__ wdec,
                      const float* __restrict__ bdec,
                      float* __restrict__ small){
  int kb = blockIdx.x;
  __shared__ float z[DZI];
  if (threadIdx.x < DZI) z[threadIdx.x] = zmu_ws[kb*DZI + threadIdx.x];
  __syncthreads();
  for (int o = threadIdx.x; o < OUT_FLAT; o += blockDim.x) {
    float s = bdec[o];
    #pragma unroll
    for (int d = 0; d < DZI; ++d) s += z[d]*wdec[o*DZI + d];
    small[(size_t)kb*OUT_FLAT + o] = (o < SS*SS) ? softplusf_(s) : sigmoidf_(s);
  }
}

// ---------------- kernel 7: uncrop + composite -----------------------------
__global__ void k_uncrop(const float* __restrict__ small,
                         const float* __restrict__ boxes,
                         float* __restrict__ out){
  int b = blockIdx.y;
  int p = blockIdx.x*blockDim.x + threadIdx.x;  // 0..65535
  int x = p >> 8, y = p & 255;
  float wreg[KOBJ];
  float sum_w = 0.f;
  #pragma unroll
  for (int k = 0; k < KOBJ; ++k) {
    int kb = k*BB + b;
    float cx = boxes[kb*4+0], cy = boxes[kb*4+1];
    float cw = boxes[kb*4+2], chh = boxes[kb*4+3];
    float sx = ((float)x - (cx - 0.5f*cw)) / cw  * (float)(SS-1);
    float sy = ((float)y - (cy - 0.5f*chh)) / chh * (float)(SS-1);
    float wv = 0.f, iv = 0.f;
    if (sx > -1.f && sx < (float)SS && sy > -1.f && sy < (float)SS) {
      const float* sm = small + (size_t)kb*OUT_FLAT;
      wv = bilin2d(sm,         sx, sy, SS, SS);   // channel 0: weight
      iv = bilin2d(sm + SS*SS, sx, sy, SS, SS);   // channel 1: image
    }
    wreg[k] = wv;
    sum_w  += wv;
    size_t pix = (size_t)kb*(WDIM*HDIM) + p;
    out[O_BMASKN + pix] = tanhf(wv);
    out[O_BIMG   + pix] = iv;
  }
  float invw = 1.f / fmaxf(sum_w, 1e-6f);
  float tm   = tanhf(sum_w);
  #pragma unroll
  for (int k = 0; k < KOBJ; ++k)
    out[O_BMASK + (size_t)(k*BB + b)*(WDIM*HDIM) + p] = tm * wreg[k] * invw;
}

// ---------------- launcher --------------------------------------------------
extern "C" void kernel_launch(void* const* d_in, const int* in_sizes, int n_in,
                              void* d_out, int out_size, void* d_ws, size_t ws_size,
                              hipStream_t stream) {
  (void)in_sizes; (void)n_in; (void)out_size; (void)ws_size;
  const float* features  = (const float*)d_in[1];
  const float* zmu       = (const float*)d_in[2];
  const float* zstd      = (const float*)d_in[3];
  const float* logit     = (const float*)d_in[4];
  const float* w_zw      = (const float*)d_in[5];
  const float* b_zw      = (const float*)d_in[6];
  const float* w_enc_mu  = (const float*)d_in[7];
  const float* b_enc_mu  = (const float*)d_in[8];
  const float* w_enc_std = (const float*)d_in[9];
  const float* b_enc_std = (const float*)d_in[10];
  const float* w_dec     = (const float*)d_in[11];
  const float* b_dec     = (const float*)d_in[12];
  float* out = (float*)d_out;
  float* ws  = (float*)d_ws;

  k_maps  <<<dim3((BB*NCELL+255)/256), 256, 0, stream>>>(logit, out);
  k_topk  <<<dim3(BB), 32, 0, stream>>>(logit, zmu, zstd, w_zw, b_zw, out, ws + W_BOXES);
  k_crop  <<<dim3(KB), 256, 0, stream>>>(features, ws + W_BOXES, ws + W_XF);
  k_enc   <<<dim3(MTILES, KCHUNKS), 32, 0, stream>>>(ws + W_XF, w_enc_mu, w_enc_std, ws + W_ZPART);
  k_post  <<<dim3((KB*DZI+255)/256), 256, 0, stream>>>(ws + W_ZPART, b_enc_mu, b_enc_std, out, ws + W_ZMU);
  k_dec   <<<dim3(KB), 256, 0, stream>>>(ws + W_ZMU, w_dec, b_dec, ws + W_SMALL);
  k_uncrop<<<dim3(WDIM*HDIM/256, BB), 256, 0, stream>>>(ws + W_SMALL, ws + W_BOXES, out);
}